// NoisyTopKGating_25220047962118
// MI455X (gfx1250) — compile-verified
//
#include <hip/hip_runtime.h>
#include <math.h>

typedef __attribute__((ext_vector_type(2))) float v2f;
typedef __attribute__((ext_vector_type(8))) float v8f;

#define B_ROWS 8192
#define D_DIM  4096
#define E_DIM  64
#define ROWS_PER_BLOCK 32   // 2 waves x 16-row M-tiles
#define THREADS 64
#define NOISE_EPS 0.01f

__global__ __launch_bounds__(THREADS)
void noisy_topk_gating_kernel(const float* __restrict__ x,
                              const float* __restrict__ gate_w,
                              const float* __restrict__ noise_gate_w,
                              const float* __restrict__ noise,
                              float* __restrict__ gates_out,
                              float* __restrict__ load_out)
{
    // +1 pad -> conflict-free per-row reads in the epilogue
    __shared__ float ldsClean[ROWS_PER_BLOCK][E_DIM + 1];
    __shared__ float ldsNoise[ROWS_PER_BLOCK][E_DIM + 1];

    const int tid  = threadIdx.x;
    const int wave = tid >> 5;        // wave32
    const int lane = tid & 31;
    const int half = lane >> 4;       // lane half selects K pair (A) / K pair (B)
    const int ln   = lane & 15;       // M index (A) / N index (B)

    const int rowBase = blockIdx.x * ROWS_PER_BLOCK + wave * 16;

    // A operand: x[rowBase+ln][k + 2*half + {0,1}]  (32-bit A 16x4 layout)
    const float* xp = x + (size_t)(rowBase + ln) * D_DIM + 2 * half;
    // B operand: W^T[k][n] = W[n][k]; n = t*16+ln striped across lanes,
    // K pairs split across lane halves (mirror of A layout)
    const float* gp[4];
    const float* qp[4];
#pragma unroll
    for (int t = 0; t < 4; ++t) {
        gp[t] = gate_w       + (size_t)(t * 16 + ln) * D_DIM + 2 * half;
        qp[t] = noise_gate_w + (size_t)(t * 16 + ln) * D_DIM + 2 * half;
    }

    v8f accC[4], accN[4];
#pragma unroll
    for (int t = 0; t < 4; ++t) { accC[t] = (v8f){}; accN[t] = (v8f){}; }

    for (int k0 = 0; k0 < D_DIM; k0 += 16) {
        // speculative prefetch of the next x slab (global_prefetch_b8)
        __builtin_prefetch(xp + k0 + 16, 0, 0);
#pragma unroll
        for (int kk = 0; kk < 16; kk += 4) {
            v2f a = *(const v2f*)(xp + k0 + kk);
#pragma unroll
            for (int t = 0; t < 4; ++t) {
                v2f bg = *(const v2f*)(gp[t] + k0 + kk);
                accC[t] = __builtin_amdgcn_wmma_f32_16x16x4_f32(
                    false, a, false, bg, (short)0, accC[t], false, false);
            }
#pragma unroll
            for (int t = 0; t < 4; ++t) {
                v2f bn = *(const v2f*)(qp[t] + k0 + kk);
                accN[t] = __builtin_amdgcn_wmma_f32_16x16x4_f32(
                    false, a, false, bn, (short)0, accN[t], false, false);
            }
        }
    }

    // C/D layout: lanes 0-15: VGPR r -> M=r; lanes 16-31: VGPR r -> M=8+r; N=ln
    const int lrBase = wave * 16 + 8 * half;
#pragma unroll
    for (int t = 0; t < 4; ++t) {
#pragma unroll
        for (int r = 0; r < 8; ++r) {
            ldsClean[lrBase + r][t * 16 + ln] = accC[t][r];
            ldsNoise[lrBase + r][t * 16 + ln] = accN[t][r];
        }
    }
    __syncthreads();

    if (tid < ROWS_PER_BLOCK) {
        // ---- gates path: noisy logits -> top-2 -> softmax -> scatter ----
        const int lr = tid;
        const size_t b = (size_t)blockIdx.x * ROWS_PER_BLOCK + lr;
        const float* nz = noise + b * E_DIM;

        float m1 = -3.402823466e38f, m2 = -3.402823466e38f;
        int   i1 = 0, i2 = 0;
#pragma unroll 4
        for (int j = 0; j < E_DIM; ++j) {
            float c = ldsClean[lr][j];
            float z = ldsNoise[lr][j];
            // numerically stable softplus
            float sp = fmaxf(z, 0.0f) + log1pf(expf(-fabsf(z)));
            float v  = c + nz[j] * sp * NOISE_EPS;
            if (v > m1)      { m2 = m1; i2 = i1; m1 = v; i1 = j; }
            else if (v > m2) { m2 = v;  i2 = j; }
        }
        // softmax over the two selected logits
        float g1 = 1.0f / (1.0f + expf(m2 - m1));
        float g2 = 1.0f - g1;

        float* gout = gates_out + b * E_DIM;
#pragma unroll 4
        for (int j = 0; j < E_DIM; ++j) {
            float g = (j == i1) ? g1 : ((j == i2) ? g2 : 0.0f);
            gout[j] = g;
        }
    } else {
        // ---- load path: softmax(clean_logits) over experts ----
        const int lr = tid - ROWS_PER_BLOCK;
        const size_t b = (size_t)blockIdx.x * ROWS_PER_BLOCK + lr;

        float mx = -3.402823466e38f;
#pragma unroll 4
        for (int j = 0; j < E_DIM; ++j)
            mx = fmaxf(mx, ldsClean[lr][j]);

        float s = 0.0f;
#pragma unroll 4
        for (int j = 0; j < E_DIM; ++j)
            s += expf(ldsClean[lr][j] - mx);
        float inv = 1.0f / s;

        float* lout = load_out + b * E_DIM;
#pragma unroll 4
        for (int j = 0; j < E_DIM; ++j)
            lout[j] = expf(ldsClean[lr][j] - mx) * inv;
    }
}

extern "C" void kernel_launch(void* const* d_in, const int* in_sizes, int n_in,
                              void* d_out, int out_size, void* d_ws, size_t ws_size,
                              hipStream_t stream) {
    const float* x  = (const float*)d_in[0];  // [B, D]
    const float* gw = (const float*)d_in[1];  // [E, D]
    const float* nw = (const float*)d_in[2];  // [E, D]
    const float* nz = (const float*)d_in[3];  // [B, E]

    float* gates = (float*)d_out;                       // [B, E]
    float* load  = gates + (size_t)B_ROWS * E_DIM;      // [B, E]

    dim3 grid(B_ROWS / ROWS_PER_BLOCK);   // 256 blocks
    dim3 block(THREADS);                  // 2 waves
    hipLaunchKernelGGL(noisy_topk_gating_kernel, grid, block, 0, stream,
                       x, gw, nw, nz, gates, load);
}